// AdapterBlock_3564822856298
// MI455X (gfx1250) — compile-verified
//
#include <hip/hip_runtime.h>
#include <math.h>

#define B_   8
#define T_   2048
#define C_   1024
#define RED_ 16
#define CH_  (C_ / RED_)   // 64

typedef __attribute__((ext_vector_type(2))) float v2f;
typedef __attribute__((ext_vector_type(8))) float v8f;

// Per-lane async DMA: copies 16 bytes from global address gptr to LDS offset lds_off, ASYNCcnt path.
// lds_off is the 32-bit LDS byte offset (low 32 bits of the generic pointer).
__device__ __forceinline__ void async_copy_b128(unsigned lds_off, const float* gptr) {
    asm volatile("global_load_async_to_lds_b128 %0, %1, off"
                 :: "v"(lds_off), "v"(gptr) : "memory");
}
__device__ __forceinline__ void async_wait0() {
    asm volatile("s_wait_asynccnt 0x0" ::: "memory");
}
__device__ __forceinline__ unsigned lds_off_of(const void* p) {
    return (unsigned)(uintptr_t)p;   // generic LDS addr: its low 32 bits are the LDS offset
}

// ---------------------------------------------------------------------------
// Kernel 1: LayerNorm statistics (mean, rstd) per (b,t) row of C=1024.
// ---------------------------------------------------------------------------
__global__ void ln_stats_kernel(const float* __restrict__ x,
                                float* __restrict__ mu,
                                float* __restrict__ rs) {
    const int row = blockIdx.x;            // b*T + t
    const int tid = threadIdx.x;           // 256 threads
    const float4* xr = (const float4*)(x + (size_t)row * C_);
    float4 v = xr[tid];                    // 256 * 4 = 1024 elements
    float s = v.x + v.y + v.z + v.w;
    float q = v.x * v.x + v.y * v.y + v.z * v.z + v.w * v.w;

    __shared__ float ss[256];
    __shared__ float sq[256];
    ss[tid] = s; sq[tid] = q;
    __syncthreads();
    for (int off = 128; off > 0; off >>= 1) {
        if (tid < off) { ss[tid] += ss[tid + off]; sq[tid] += sq[tid + off]; }
        __syncthreads();
    }
    if (tid == 0) {
        float m   = ss[0] * (1.0f / C_);
        float var = sq[0] * (1.0f / C_) - m * m;
        mu[row] = m;
        rs[row] = rsqrtf(var + 1e-5f);
    }
}

// ---------------------------------------------------------------------------
// Kernel 2: transpose [B,T,C] -> xn[B,C,T] with async-to-LDS tile copy; the
// LayerNorm affine is applied on the write phase (mu/rs coalesced over t,
// ln_w/ln_b uniform per iteration -> scalar loads).
// blockDim = (32, 8); grid = (T/32, C/32, B)
// ---------------------------------------------------------------------------
__global__ void norm_transpose_kernel(const float* __restrict__ x,
                                      const float* __restrict__ ln_w,
                                      const float* __restrict__ ln_b,
                                      const float* __restrict__ mu,
                                      const float* __restrict__ rs,
                                      float* __restrict__ xn) {
    __shared__ __align__(16) float tile[32][36];   // [t_local][c_local]
    const int b  = blockIdx.z;
    const int t0 = blockIdx.x * 32;
    const int c0 = blockIdx.y * 32;
    const int tx = threadIdx.x;                    // 0..31
    const int ty = threadIdx.y;                    // 0..7
    const int tid = ty * 32 + tx;                  // 0..255

    // Async raw copy of the 32x32 x tile: lane -> (row tl, 16B segment seg).
    {
        const int tl  = tid >> 3;
        const int seg = (tid & 7) * 4;
        const float* g = x + ((size_t)(b * T_ + t0 + tl)) * C_ + c0 + seg;
        async_copy_b128(lds_off_of(&tile[tl][seg]), g);
    }
    async_wait0();
    __syncthreads();

    const float m = mu[b * T_ + t0 + tx];
    const float r = rs[b * T_ + t0 + tx];
#pragma unroll
    for (int j = 0; j < 4; ++j) {
        int cl = ty + 8 * j;
        float w  = ln_w[c0 + cl];                  // uniform -> scalar
        float bb = ln_b[c0 + cl];
        float v  = tile[tx][cl];
        // coalesced over t
        xn[((size_t)(b * C_ + c0 + cl)) * T_ + t0 + tx] = (v - m) * r * w + bb;
    }
}

// ---------------------------------------------------------------------------
// Kernel 3: fused depthwise conv3 -> ReLU -> conv5 -> conv3 along T, per
// (b,c) row, plus time-mean pooling. Row staged in LDS via async DMA with a
// zero halo; output overwrites the input row IN PLACE (all global reads
// complete at the async wait before any store). blockDim = 256; grid = B*C
// ---------------------------------------------------------------------------
__global__ void dwconv_chain_kernel(const float* __restrict__ xn,
                                    const float* __restrict__ w1,
                                    const float* __restrict__ w2,
                                    const float* __restrict__ w3,
                                    float* __restrict__ y,
                                    float* __restrict__ pooled) {
    const int bc  = blockIdx.x;        // b*C + c
    const int c   = bc & (C_ - 1);
    const int tid = threadIdx.x;       // 256

    __shared__ __align__(16) float s[T_ + 8];   // +4 halo each side (zeros)
    const float* row = xn + (size_t)bc * T_;
    if (tid < 4) { s[tid] = 0.0f; s[T_ + 4 + tid] = 0.0f; }
    {
        const int o = tid * 4;                  // two b128 segments per lane
        async_copy_b128(lds_off_of(&s[4 + o]),        row + o);
        async_copy_b128(lds_off_of(&s[4 + 1024 + o]), row + 1024 + o);
    }
    async_wait0();
    __syncthreads();

    const float a0 = w1[c * 3 + 0], a1 = w1[c * 3 + 1], a2 = w1[c * 3 + 2];
    const float b0 = w2[c * 5 + 0], b1 = w2[c * 5 + 1], b2 = w2[c * 5 + 2],
                b3 = w2[c * 5 + 3], b4 = w2[c * 5 + 4];
    const float d0 = w3[c * 3 + 0], d1 = w3[c * 3 + 1], d2 = w3[c * 3 + 2];

    const int tb = tid * 8;            // this thread's 8 output positions
    float xv[16];                      // xn[tb-4 .. tb+11] (s is +4 shifted)
#pragma unroll
    for (int i = 0; i < 16; ++i) xv[i] = s[tb + i];

    // stage 1: conv3 + relu; SAME padding applies per-stage, so taps outside
    // [0,T) must be zeroed, not recomputed from padded x.
    float o1[14];                      // t = tb-3 .. tb+10
#pragma unroll
    for (int i = 0; i < 14; ++i) {
        int t = tb - 3 + i;
        float v = fmaxf(a0 * xv[i] + a1 * xv[i + 1] + a2 * xv[i + 2], 0.0f);
        o1[i] = (t >= 0 && t < T_) ? v : 0.0f;
    }
    // stage 2: conv5
    float o2[10];                      // t = tb-1 .. tb+8
#pragma unroll
    for (int j = 0; j < 10; ++j) {
        int t = tb - 1 + j;
        float v = b0 * o1[j] + b1 * o1[j + 1] + b2 * o1[j + 2] +
                  b3 * o1[j + 3] + b4 * o1[j + 4];
        o2[j] = (t >= 0 && t < T_) ? v : 0.0f;
    }
    // stage 3: conv3 + running pool sum
    float o3[8];
    float acc = 0.0f;
#pragma unroll
    for (int k = 0; k < 8; ++k) {
        o3[k] = d0 * o2[k] + d1 * o2[k + 1] + d2 * o2[k + 2];
        acc += o3[k];
    }

    float4* yo = (float4*)(y + (size_t)bc * T_ + tb);
    yo[0] = make_float4(o3[0], o3[1], o3[2], o3[3]);
    yo[1] = make_float4(o3[4], o3[5], o3[6], o3[7]);

    __shared__ float red[256];
    red[tid] = acc;
    __syncthreads();
    for (int off = 128; off > 0; off >>= 1) {
        if (tid < off) red[tid] += red[tid + off];
        __syncthreads();
    }
    if (tid == 0) pooled[bc] = red[0] * (1.0f / T_);
}

// ---------------------------------------------------------------------------
// Kernel 4: SE block with fp32 WMMA (V_WMMA_F32_16X16X4_F32).
//   H[16x64]   = relu( pooled[16x1024] * fc1^T[1024x64] )   (M padded 8->16)
//   G[16x1024] = sigmoid( H * fc2^T[64x1024] )
// One block, 4 waves of 32 (wave32). 32-bit A operand: lane = M, half-wave
// selects K pair {0,1}/{2,3}; B operand mirrors with lane = N.
// ---------------------------------------------------------------------------
__global__ void se_wmma_kernel(const float* __restrict__ pooled,
                               const float* __restrict__ fc1,
                               const float* __restrict__ fc2,
                               float* __restrict__ gate) {
    __shared__ float Hs[16][CH_ + 1];
    const int tid  = threadIdx.x;        // 128
    const int wave = tid >> 5;           // 0..3
    const int lane = tid & 31;
    const int m    = lane & 15;
    const int koff = (lane >> 4) << 1;   // 0 or 2
    const float mmask = (m < B_) ? 1.0f : 0.0f;
    const int msafe = m & (B_ - 1);      // clamp row index, mask by multiply

    // ---- GEMM1: hidden = pooled * fc1^T, each wave owns one 16-wide N tile
    v8f acc = {};
    const int n1 = wave * 16 + m;        // hidden channel (0..63) = fc1 row
    for (int k0 = 0; k0 < C_; k0 += 4) {
        v2f a, bm;
        a.x  = pooled[msafe * C_ + k0 + koff]     * mmask;
        a.y  = pooled[msafe * C_ + k0 + koff + 1] * mmask;
        bm.x = fc1[n1 * C_ + k0 + koff];
        bm.y = fc1[n1 * C_ + k0 + koff + 1];
        acc = __builtin_amdgcn_wmma_f32_16x16x4_f32(
            false, a, false, bm, (short)0, acc, false, false);
    }
#pragma unroll
    for (int i = 0; i < 8; ++i) {
        int rrow = i + ((lane < 16) ? 0 : 8);   // D row layout
        Hs[rrow][wave * 16 + m] = fmaxf(acc[i], 0.0f);
    }
    __syncthreads();

    // ---- GEMM2: gate = sigmoid(H * fc2^T), 64 N tiles split across 4 waves
    for (int nt = wave; nt < C_ / 16; nt += 4) {
        v8f acc2 = {};
        const int n2 = nt * 16 + m;      // output channel = fc2 row
#pragma unroll
        for (int k0 = 0; k0 < CH_; k0 += 4) {
            v2f a, bm;
            a.x  = Hs[m][k0 + koff];
            a.y  = Hs[m][k0 + koff + 1];
            bm.x = fc2[n2 * CH_ + k0 + koff];
            bm.y = fc2[n2 * CH_ + k0 + koff + 1];
            acc2 = __builtin_amdgcn_wmma_f32_16x16x4_f32(
                false, a, false, bm, (short)0, acc2, false, false);
        }
#pragma unroll
        for (int i = 0; i < 8; ++i) {
            int rrow = i + ((lane < 16) ? 0 : 8);
            if (rrow < B_) {
                float g = 1.0f / (1.0f + __expf(-acc2[i]));
                gate[rrow * C_ + nt * 16 + m] = g;
            }
        }
    }
}

// ---------------------------------------------------------------------------
// Kernel 5: transpose back to [B,T,C] with async-to-LDS tile copy; gate
// multiply + residual add applied on the write phase (gate coalesced over c).
// blockDim = (32, 8); grid = (C/32, T/32, B)
// ---------------------------------------------------------------------------
__global__ void gate_transpose_residual_kernel(const float* __restrict__ y,
                                               const float* __restrict__ gate,
                                               const float* __restrict__ resid,
                                               float* __restrict__ out) {
    __shared__ __align__(16) float tile[32][36];   // [c_local][t_local]
    const int b  = blockIdx.z;
    const int c0 = blockIdx.x * 32;
    const int t0 = blockIdx.y * 32;
    const int tx = threadIdx.x;
    const int ty = threadIdx.y;
    const int tid = ty * 32 + tx;

    // Async raw copy of the 32x32 y tile (rows contiguous along t).
    {
        const int cl  = tid >> 3;
        const int seg = (tid & 7) * 4;
        const float* g = y + ((size_t)(b * C_ + c0 + cl)) * T_ + t0 + seg;
        async_copy_b128(lds_off_of(&tile[cl][seg]), g);
    }
    async_wait0();
    __syncthreads();

    const float gt = gate[b * C_ + c0 + tx];       // coalesced over c
#pragma unroll
    for (int j = 0; j < 4; ++j) {
        int tl = ty + 8 * j;
        size_t idx = ((size_t)(b * T_ + t0 + tl)) * C_ + c0 + tx;  // coalesced over c
        out[idx] = resid[idx] + tile[tx][tl] * gt;
    }
}

// ---------------------------------------------------------------------------
extern "C" void kernel_launch(void* const* d_in, const int* in_sizes, int n_in,
                              void* d_out, int out_size, void* d_ws, size_t ws_size,
                              hipStream_t stream) {
    const float* x    = (const float*)d_in[0];
    const float* res  = (const float*)d_in[1];
    const float* ln_w = (const float*)d_in[2];
    const float* ln_b = (const float*)d_in[3];
    const float* w1   = (const float*)d_in[4];
    const float* w2   = (const float*)d_in[5];
    const float* w3   = (const float*)d_in[6];
    const float* fc1  = (const float*)d_in[7];
    const float* fc2  = (const float*)d_in[8];
    float* out = (float*)d_out;

    // Workspace layout (floats). xn is reused in place for the conv output
    // (kernel 3 fully stages its row in LDS before writing), so we only need
    // one [B,C,T] buffer: ~64 MB + small tails.
    float* ws = (float*)d_ws;
    const size_t N = (size_t)B_ * C_ * T_;
    float* xn     = ws;                      // [B,C,T] normalized / conv out
    float* pooled = ws + N;                  // [B,C]
    float* gate   = pooled + (size_t)B_ * C_;// [B,C]
    float* mu     = gate + (size_t)B_ * C_;  // [B*T]
    float* rs     = mu + (size_t)B_ * T_;    // [B*T]

    ln_stats_kernel<<<B_ * T_, 256, 0, stream>>>(x, mu, rs);

    dim3 blk(32, 8);
    norm_transpose_kernel<<<dim3(T_ / 32, C_ / 32, B_), blk, 0, stream>>>(
        x, ln_w, ln_b, mu, rs, xn);

    dwconv_chain_kernel<<<B_ * C_, 256, 0, stream>>>(xn, w1, w2, w3, xn, pooled);

    se_wmma_kernel<<<1, 128, 0, stream>>>(pooled, fc1, fc2, gate);

    gate_transpose_residual_kernel<<<dim3(C_ / 32, T_ / 32, B_), blk, 0, stream>>>(
        xn, gate, res, out);
}